// GraphSAGE_13889924235783
// MI455X (gfx1250) — compile-verified
//
#include <hip/hip_runtime.h>
#include <hip/hip_bf16.h>

// ---------------------------------------------------------------------------
// GraphSAGE (2x SAGEConv mean-aggr + global mean pool + linear classifier)
// MI455X / gfx1250: wave32, WMMA f32 16x16x4 for full-precision GEMMs.
// GEMM blocks cover 80-row M tiles (5x16) so each B (weight) fetch feeds
// 5 WMMAs -> 5x less L2 weight traffic than a 16-row tile.
// ---------------------------------------------------------------------------

#define N_NODES   50000
#define N_EDGES   800000
#define D_FEAT    256
#define N_CLASSES 32
#define N_GRAPHS  64
#define MT        5           // M-subtiles (16 rows each) per block: 80 rows
#define M_TILE    (16 * MT)

typedef __attribute__((ext_vector_type(2))) float v2f;
typedef __attribute__((ext_vector_type(8))) float v8f;

// ---------------------------------------------------------------------------
// In-degree count (as float) via L2 atomics: cnt[dst[e]] += 1
// ---------------------------------------------------------------------------
__global__ void deg_count_kernel(const int* __restrict__ dst,
                                 float* __restrict__ cnt) {
    int e = blockIdx.x * blockDim.x + threadIdx.x;
    if (e < N_EDGES) atomicAdd(&cnt[dst[e]], 1.0f);
}

// cnt[n] -> 1 / max(cnt[n], 1)   (in place)
__global__ void deg_invert_kernel(float* __restrict__ cnt) {
    int n = blockIdx.x * blockDim.x + threadIdx.x;
    if (n < N_NODES) cnt[n] = 1.0f / fmaxf(cnt[n], 1.0f);
}

// ---------------------------------------------------------------------------
// Edge scatter: agg[dst] += feat[src].  One thread per (edge, 4-float chunk):
// float4 gather (features are L2-resident: 51.2 MB < 192 MB L2) followed by
// 4 global_atomic_add_f32 into the L2 atomic units.
// ---------------------------------------------------------------------------
__global__ void scatter_sum_kernel(const float* __restrict__ feat,
                                   const int* __restrict__ src,
                                   const int* __restrict__ dst,
                                   float* __restrict__ agg) {
    long long t = (long long)blockIdx.x * blockDim.x + threadIdx.x;
    int e = (int)(t >> 6);          // 64 chunks of 4 floats per edge
    int c = ((int)t & 63) * 4;
    if (e < N_EDGES) {
        int s = src[e], d = dst[e];
        const float4 v = *(const float4*)(feat + (size_t)s * D_FEAT + c);
        float* p = agg + (size_t)d * D_FEAT + c;
        atomicAdd(p + 0, v.x);
        atomicAdd(p + 1, v.y);
        atomicAdd(p + 2, v.z);
        atomicAdd(p + 3, v.w);
    }
}

// ---------------------------------------------------------------------------
// Fused SAGE GEMM using V_WMMA_F32_16X16X4_F32 (full f32 precision):
//   out[m,:] = relu( (agg[m,:]*inv[m]) @ Wl + x[m,:] @ Wr + b )
//
// Block: 256 threads = 8 waves, covers an 80-row M tile over all 256 columns.
// Wave w owns columns [w*32, w*32+32): 2 N-subtiles x 5 M-subtiles =
// 10 f32 16x16 accumulators (80 VGPRs).  Per k-step the 4 B dwords are
// loaded once and reused against 5 A float2 loads (10 WMMAs).
//
// Operand layouts (ISA 7.12.2, wave32):
//   A 16x4: lanes 0-15 hold K={k,k+1}, lanes 16-31 K={k+2,k+3} (float2 load)
//   B 4x16: VGPR0 = rows K / K+2, VGPR1 = rows K+1 / K+3 (mirrors A split)
//   C/D   : VGPR v -> row v + 8*(lane>=16), col = lane&15
//
// In-place out==agg is safe: each block reads only its own 80 rows, and a
// __syncthreads() separates all reads from the stores.
// ---------------------------------------------------------------------------
__global__ __launch_bounds__(256)
void sage_gemm_kernel(const float* __restrict__ agg,
                      const float* __restrict__ inv,
                      const float* __restrict__ xin,
                      const float* __restrict__ Wl,
                      const float* __restrict__ Wr,
                      const float* __restrict__ bias,
                      float* __restrict__ out) {
    const int lane = threadIdx.x & 31;
    const int wave = threadIdx.x >> 5;
    const int half = lane >> 4;      // 0: lanes 0-15, 1: lanes 16-31
    const int l    = lane & 15;
    const int m0   = blockIdx.x * M_TILE;
    const int n0   = wave * 32;

    // Per-lane A row bases (row = m0 + i*16 + l); i handled via immediate
    // offsets of i*16*256 floats (16 KB, fits 24-bit IOFFSET).
    const float* abase = agg + (size_t)(m0 + l) * D_FEAT;
    const float* xbase = xin + (size_t)(m0 + l) * D_FEAT;

    float scale[MT];
    #pragma unroll
    for (int i = 0; i < MT; ++i) scale[i] = inv[m0 + i * 16 + l];

    v8f c0[MT];   // columns n0 .. n0+15
    v8f c1[MT];   // columns n0+16 .. n0+31
    #pragma unroll
    for (int i = 0; i < MT; ++i) { c0[i] = (v8f){}; c1[i] = (v8f){}; }

    // ---- agg (mean-normalized) @ Wl ----
    for (int k = 0; k < D_FEAT; k += 4) {
        const int kk = k + half * 2;
        const float* w = Wl + (size_t)kk * D_FEAT + n0 + l;
        v2f b0; b0.x = w[0];  b0.y = w[D_FEAT];        // rows kk, kk+1
        v2f b1; b1.x = w[16]; b1.y = w[D_FEAT + 16];
        #pragma unroll
        for (int i = 0; i < MT; ++i) {
            v2f a = *(const v2f*)(abase + i * 16 * D_FEAT + kk);
            a.x *= scale[i]; a.y *= scale[i];
            c0[i] = __builtin_amdgcn_wmma_f32_16x16x4_f32(
                        false, a, false, b0, (short)0, c0[i], false, false);
            c1[i] = __builtin_amdgcn_wmma_f32_16x16x4_f32(
                        false, a, false, b1, (short)0, c1[i], false, false);
        }
    }

    // ---- x @ Wr ----
    for (int k = 0; k < D_FEAT; k += 4) {
        const int kk = k + half * 2;
        const float* w = Wr + (size_t)kk * D_FEAT + n0 + l;
        v2f b0; b0.x = w[0];  b0.y = w[D_FEAT];
        v2f b1; b1.x = w[16]; b1.y = w[D_FEAT + 16];
        #pragma unroll
        for (int i = 0; i < MT; ++i) {
            v2f a = *(const v2f*)(xbase + i * 16 * D_FEAT + kk);
            c0[i] = __builtin_amdgcn_wmma_f32_16x16x4_f32(
                        false, a, false, b0, (short)0, c0[i], false, false);
            c1[i] = __builtin_amdgcn_wmma_f32_16x16x4_f32(
                        false, a, false, b1, (short)0, c1[i], false, false);
        }
    }

    __syncthreads();   // all reads of agg rows done -> safe for out==agg

    const float bn0 = bias[n0 + l];
    const float bn1 = bias[n0 + 16 + l];
    #pragma unroll
    for (int i = 0; i < MT; ++i) {
        #pragma unroll
        for (int v = 0; v < 8; ++v) {
            const int r = m0 + i * 16 + v + half * 8;
            float o0 = c0[i][v] + bn0; o0 = o0 > 0.0f ? o0 : 0.0f;
            float o1 = c1[i][v] + bn1; o1 = o1 > 0.0f ? o1 : 0.0f;
            out[(size_t)r * D_FEAT + n0 + l]      = o0;
            out[(size_t)r * D_FEAT + n0 + 16 + l] = o1;
        }
    }
}

// ---------------------------------------------------------------------------
// Global mean pool: gsum[batch[n]] += h[n], gcnt[batch[n]] += 1
// ---------------------------------------------------------------------------
__global__ void pool_kernel(const float* __restrict__ h,
                            const int* __restrict__ batch,
                            float* __restrict__ gsum,
                            float* __restrict__ gcnt) {
    long long t = (long long)blockIdx.x * blockDim.x + threadIdx.x;
    int n = (int)(t >> 6);
    int c = ((int)t & 63) * 4;
    if (n < N_NODES) {
        int g = batch[n];
        const float4 v = *(const float4*)(h + (size_t)n * D_FEAT + c);
        float* p = gsum + (size_t)g * D_FEAT + c;
        atomicAdd(p + 0, v.x);
        atomicAdd(p + 1, v.y);
        atomicAdd(p + 2, v.z);
        atomicAdd(p + 3, v.w);
        if (c == 0) atomicAdd(&gcnt[g], 1.0f);
    }
}

// ---------------------------------------------------------------------------
// Classifier: out[g,c] = bc[c] + (gsum[g,:]/max(gcnt[g],1)) @ Wc[:,c]
// 64*32 = 2048 outputs, 1 MFLOP total -> trivial scalar kernel.
// ---------------------------------------------------------------------------
__global__ void classify_kernel(const float* __restrict__ gsum,
                                const float* __restrict__ gcnt,
                                const float* __restrict__ Wc,
                                const float* __restrict__ bc,
                                float* __restrict__ out) {
    int t = blockIdx.x * blockDim.x + threadIdx.x;
    if (t >= N_GRAPHS * N_CLASSES) return;
    int g = t >> 5;          // /32
    int c = t & 31;
    float invc = 1.0f / fmaxf(gcnt[g], 1.0f);
    float acc = bc[c];
    const float* gs = gsum + (size_t)g * D_FEAT;
    for (int k = 0; k < D_FEAT; ++k)
        acc = fmaf(gs[k] * invc, Wc[(size_t)k * N_CLASSES + c], acc);
    out[t] = acc;
}

// ---------------------------------------------------------------------------
// Host launcher
// Inputs: 0:x 1:edge_index 2:batch 3:W1_l 4:b1 5:W1_r 6:W2_l 7:b2 8:W2_r
//         9:Wc 10:bc
// ---------------------------------------------------------------------------
extern "C" void kernel_launch(void* const* d_in, const int* in_sizes, int n_in,
                              void* d_out, int out_size, void* d_ws, size_t ws_size,
                              hipStream_t stream) {
    const float* x    = (const float*)d_in[0];
    const int*   src  = (const int*)d_in[1];             // edge_index row 0
    const int*   dst  = ((const int*)d_in[1]) + N_EDGES; // edge_index row 1
    const int*   batch= (const int*)d_in[2];
    const float* W1l  = (const float*)d_in[3];
    const float* b1   = (const float*)d_in[4];
    const float* W1r  = (const float*)d_in[5];
    const float* W2l  = (const float*)d_in[6];
    const float* b2   = (const float*)d_in[7];
    const float* W2r  = (const float*)d_in[8];
    const float* Wc   = (const float*)d_in[9];
    const float* bc   = (const float*)d_in[10];
    float* out = (float*)d_out;

    const size_t featBytes = (size_t)N_NODES * D_FEAT * sizeof(float); // 51.2 MB
    char* ws = (char*)d_ws;
    float* agg  = (float*)(ws);                         // agg1 / agg2 / h2
    float* h1   = (float*)(ws + featBytes);             // h1
    float* inv  = (float*)(ws + 2 * featBytes);         // degree inverse
    float* gsum = (float*)(ws + 2 * featBytes + 256 * 1024);
    float* gcnt = (float*)(ws + 2 * featBytes + 256 * 1024 + 64 * 1024);

    const int edgeBlocks    = (N_EDGES + 255) / 256;                  // 3125
    const int nodeBlocks    = (N_NODES + 255) / 256;                  // 196
    const int scatterBlocks = (int)(((long long)N_EDGES * 64) / 256); // 200000
    const int poolBlocks    = (int)(((long long)N_NODES * 64) / 256); // 12500
    const int gemmBlocks    = N_NODES / M_TILE;                       // 625

    // ---- degree -> 1/max(deg,1) ----
    hipMemsetAsync(inv, 0, (size_t)N_NODES * sizeof(float), stream);
    deg_count_kernel<<<edgeBlocks, 256, 0, stream>>>(dst, inv);
    deg_invert_kernel<<<nodeBlocks, 256, 0, stream>>>(inv);

    // ---- layer 1 ----
    hipMemsetAsync(agg, 0, featBytes, stream);
    scatter_sum_kernel<<<scatterBlocks, 256, 0, stream>>>(x, src, dst, agg);
    sage_gemm_kernel<<<gemmBlocks, 256, 0, stream>>>(agg, inv, x, W1l, W1r, b1, h1);

    // ---- layer 2 (GEMM output written in place over agg) ----
    hipMemsetAsync(agg, 0, featBytes, stream);
    scatter_sum_kernel<<<scatterBlocks, 256, 0, stream>>>(h1, src, dst, agg);
    sage_gemm_kernel<<<gemmBlocks, 256, 0, stream>>>(agg, inv, h1, W2l, W2r, b2, agg);

    // ---- global mean pool + classifier ----
    hipMemsetAsync(gsum, 0, (size_t)N_GRAPHS * D_FEAT * sizeof(float), stream);
    hipMemsetAsync(gcnt, 0, (size_t)N_GRAPHS * sizeof(float), stream);
    pool_kernel<<<poolBlocks, 256, 0, stream>>>(agg, batch, gsum, gcnt);
    classify_kernel<<<(N_GRAPHS * N_CLASSES + 255) / 256, 256, 0, stream>>>(
        gsum, gcnt, Wc, bc, out);
}